// GCN2_3745211482882
// MI455X (gfx1250) — compile-verified
//
#include <hip/hip_runtime.h>
#include <math.h>

// GCN 3-layer pipeline for MI455X (gfx1250), wave32.
// Layer sizes: 1 -> 100 -> 10 -> 1, N=50000 nodes, E=800000 edges.
// The only non-trivial GEMM (N,100)@(100,10) runs on V_WMMA_F32_16X16X4_F32
// (exact f32 matrix core path). Layer-0's (N,100) activation is generated
// on the fly inside the WMMA kernel (it is rank-1 + bias + leaky_relu),
// avoiding a 20 MB intermediate entirely. W1 is zero-padded to 16 columns in
// LDS so the WMMA operand feed has no divergent loads (no saveexec churn).

#define NN 50000

typedef __attribute__((ext_vector_type(2))) float v2f;
typedef __attribute__((ext_vector_type(8))) float v8f;

__global__ void k_zero(float* __restrict__ p, int n) {
  int i = blockIdx.x * blockDim.x + threadIdx.x;
  if (i < n) p[i] = 0.0f;
}

// Degree counting: deg_out from src, deg_in from dst.
__global__ void k_degrees(const int* __restrict__ src, const int* __restrict__ dst,
                          int E, float* __restrict__ deg_out, float* __restrict__ deg_in) {
  int e = blockIdx.x * blockDim.x + threadIdx.x;
  if (e < E) {
    atomicAdd(&deg_out[src[e]], 1.0f);
    atomicAdd(&deg_in[dst[e]], 1.0f);
  }
}

// In-place: d -> (max(d,1))^-1/2 for both degree buffers.
__global__ void k_invsqrt(float* __restrict__ a, float* __restrict__ b, int n) {
  int i = blockIdx.x * blockDim.x + threadIdx.x;
  if (i < n) {
    float da = a[i]; da = da > 1.0f ? da : 1.0f; a[i] = 1.0f / sqrtf(da);
    float db = b[i]; db = db > 1.0f ? db : 1.0f; b[i] = 1.0f / sqrtf(db);
  }
}

// Layer-0 aggregation (features are scalar): agg0[dst] += x[src]*inv_out[src]
__global__ void k_agg0(const int* __restrict__ src, const int* __restrict__ dst, int E,
                       const float* __restrict__ x, const float* __restrict__ inv_out,
                       float* __restrict__ agg0) {
  int e = blockIdx.x * blockDim.x + threadIdx.x;
  if (e < E) {
    int s = src[e];
    atomicAdd(&agg0[dst[e]], x[s] * inv_out[s]);
  }
}

// Fused: h0 = leaky_relu(agg0[i]*inv_in[i]*W0 + b0)  (row generated on the fly)
//        h1mm = (inv_out[i] * h0) @ W1               (f32 WMMA, K=100 = 25x4)
// One wave per 16-row tile. N = 3125*16 exactly, no remainder.
__global__ void k_l1_wmma(const float* __restrict__ agg0,
                          const float* __restrict__ inv_in,
                          const float* __restrict__ inv_out,
                          const float* __restrict__ W0,   // 100
                          const float* __restrict__ b0,   // 100
                          const float* __restrict__ W1,   // 100 x 10 row-major
                          float* __restrict__ h1mm,       // N x 16 (cols 10..15 = 0)
                          int n_tiles) {
  __shared__ float sW0[100];
  __shared__ float sb0[100];
  __shared__ float sW1p[100 * 16];        // W1 zero-padded to 16 cols
  for (int t = threadIdx.x; t < 100; t += blockDim.x) { sW0[t] = W0[t]; sb0[t] = b0[t]; }
  for (int t = threadIdx.x; t < 1600; t += blockDim.x) {
    int row = t >> 4, c = t & 15;
    sW1p[t] = (c < 10) ? W1[row * 10 + c] : 0.0f;
  }
  __syncthreads();

  const int wave = threadIdx.x >> 5;
  const int lane = threadIdx.x & 31;
  const int tile = blockIdx.x * (blockDim.x >> 5) + wave;
  if (tile >= n_tiles) return;            // wave-uniform: EXEC stays all-ones
  const int i0   = tile * 16;
  const int half = lane >> 4;             // lane group selects K pair
  const int m    = lane & 15;             // A row owned by this lane
  const int col  = lane & 15;             // B/C column owned by this lane

  // scalar driving the whole A row: agg0*inv_in (layer-0 dest-side norm)
  const float s = agg0[i0 + m] * inv_in[i0 + m];

  v8f c = {};
#pragma unroll
  for (int kk = 0; kk < 25; ++kk) {
    const int k0 = kk * 4 + half * 2;     // K indices {k0, k0+1} for this lane half
    // A tile (16x4 f32): VGPR0 -> K=k0, VGPR1 -> K=k0+1 (per ISA layout table)
    float a0 = fmaf(s, sW0[k0], sb0[k0]);
    a0 = a0 >= 0.0f ? a0 : 0.01f * a0;    // leaky_relu(0.01)
    float a1 = fmaf(s, sW0[k0 + 1], sb0[k0 + 1]);
    a1 = a1 >= 0.0f ? a1 : 0.01f * a1;
    v2f a = {a0, a1};
    // B tile (4x16 f32): row k0 in VGPR0, row k0+1 in VGPR1; pre-padded, no
    // divergence, bank-conflict-free (stride-16 rows over 64 banks).
    v2f b = {sW1p[k0 * 16 + col], sW1p[(k0 + 1) * 16 + col]};
    // D = A x B + C, exact fp32
    c = __builtin_amdgcn_wmma_f32_16x16x4_f32(false, a, false, b, (short)0, c,
                                              false, false);
  }

  // C layout: VGPR v -> row (v + 8*half), col = lane&15. Fold in source-side
  // norm for layer 1 (inv_out applied post-GEMM; row scaling commutes).
#pragma unroll
  for (int v = 0; v < 8; ++v) {
    const int i = i0 + v + 8 * half;
    h1mm[i * 16 + col] = c[v] * inv_out[i];
  }
}

// Layer-1 aggregation: 10 features/edge, 10 consecutive threads share one edge
// so the h1mm gather is a contiguous 40B read per edge.
__global__ void k_agg1(const int* __restrict__ src, const int* __restrict__ dst, int E,
                       const float* __restrict__ h1mm, float* __restrict__ agg1) {
  int t = blockIdx.x * blockDim.x + threadIdx.x;
  if (t < E * 10) {
    int e = t / 10;
    int k = t - e * 10;
    atomicAdd(&agg1[dst[e] * 10 + k], h1mm[src[e] * 16 + k]);
  }
}

// Fused layer-1 epilogue + layer-2 GEMV:
// h1 = relu(agg1*inv_in + b1); h2pre = (h1 . W2) * inv_out
__global__ void k_l2(const float* __restrict__ agg1, const float* __restrict__ inv_in,
                     const float* __restrict__ inv_out, const float* __restrict__ b1,
                     const float* __restrict__ W2, float* __restrict__ h2, int n) {
  int i = blockIdx.x * blockDim.x + threadIdx.x;
  if (i >= n) return;
  float si = inv_in[i];
  float acc = 0.0f;
#pragma unroll
  for (int k = 0; k < 10; ++k) {
    float v = fmaf(agg1[i * 10 + k], si, b1[k]);
    v = v > 0.0f ? v : 0.0f;
    acc = fmaf(v, W2[k], acc);
  }
  h2[i] = acc * inv_out[i];
}

// Layer-2 aggregation (scalar features).
__global__ void k_agg2(const int* __restrict__ src, const int* __restrict__ dst, int E,
                       const float* __restrict__ h2, float* __restrict__ agg2) {
  int e = blockIdx.x * blockDim.x + threadIdx.x;
  if (e < E) atomicAdd(&agg2[dst[e]], h2[src[e]]);
}

__global__ void k_out(const float* __restrict__ agg2, const float* __restrict__ inv_in,
                      const float* __restrict__ b2, float* __restrict__ out, int n) {
  int i = blockIdx.x * blockDim.x + threadIdx.x;
  if (i < n) {
    float v = fmaf(agg2[i], inv_in[i], b2[0]);
    out[i] = v > 0.0f ? v : 0.0f;
  }
}

extern "C" void kernel_launch(void* const* d_in, const int* in_sizes, int n_in,
                              void* d_out, int out_size, void* d_ws, size_t ws_size,
                              hipStream_t stream) {
  const float* in_feat = (const float*)d_in[0];
  const int*   edge    = (const int*)d_in[1];
  const float* W0      = (const float*)d_in[2];
  const float* b0      = (const float*)d_in[3];
  const float* W1      = (const float*)d_in[4];
  const float* b1      = (const float*)d_in[5];
  const float* W2      = (const float*)d_in[6];
  const float* b2      = (const float*)d_in[7];
  float* out = (float*)d_out;

  const int N = NN;
  const int E = in_sizes[1] / 2;
  const int* src = edge;
  const int* dst = edge + E;

  // Workspace layout (floats):
  float* ws      = (float*)d_ws;
  float* inv_out = ws;             // N   (degree counter -> inv sqrt)
  float* inv_in  = ws + N;         // N
  float* agg0    = ws + 2 * N;     // N
  float* agg1    = ws + 3 * N;     // 10N
  float* agg2    = ws + 13 * N;    // N
  float* h1mm    = ws + 14 * N;    // 16N (fully overwritten each call)
  float* h2      = ws + 30 * N;    // N   (fully overwritten each call)

  const int B = 256;
  // zero the accumulator region (first 14N floats)
  int nz = 14 * N;
  k_zero<<<(nz + B - 1) / B, B, 0, stream>>>(ws, nz);

  k_degrees<<<(E + B - 1) / B, B, 0, stream>>>(src, dst, E, inv_out, inv_in);
  k_invsqrt<<<(N + B - 1) / B, B, 0, stream>>>(inv_out, inv_in, N);
  k_agg0<<<(E + B - 1) / B, B, 0, stream>>>(src, dst, E, in_feat, inv_out, agg0);

  const int n_tiles = (N + 15) / 16;           // 3125 (exact)
  const int waves_per_block = B / 32;          // 8 tiles per block
  k_l1_wmma<<<(n_tiles + waves_per_block - 1) / waves_per_block, B, 0, stream>>>(
      agg0, inv_in, inv_out, W0, b0, W1, h1mm, n_tiles);

  int total1 = E * 10;
  k_agg1<<<(total1 + B - 1) / B, B, 0, stream>>>(src, dst, E, h1mm, agg1);
  k_l2<<<(N + B - 1) / B, B, 0, stream>>>(agg1, inv_in, inv_out, b1, W2, h2, N);
  k_agg2<<<(E + B - 1) / B, B, 0, stream>>>(src, dst, E, h2, agg2);
  k_out<<<(N + B - 1) / B, B, 0, stream>>>(agg2, inv_in, b2, out, N);
}